// IWT_72267119723414
// MI455X (gfx1250) — compile-verified
//
#include <hip/hip_runtime.h>
#include <stdint.h>

// 3D inverse wavelet (Haar butterfly) — streaming kernel using the CDNA5
// async-copy engine. Traffic-bound: 453 MB @ 23.3 TB/s -> ~19.4 us floor.
// Inputs are staged memory->LDS with GLOBAL_LOAD_ASYNC_TO_LDS_B128 (ASYNCcnt),
// read back with ds_load_b128, butterflied in registers, and written with
// non-temporal B128 stores (write-once stream > 192 MB L2).

typedef float f4 __attribute__((ext_vector_type(4)));

constexpr int B = 2, C = 32, D = 48, H = 48, W = 48;
constexpr int CSTRIDE = D * H * W;        // input per-channel stride (110592)
constexpr int GSTRIDE = C * CSTRIDE;      // input wavelet-group stride
constexpr int BSTRIDE = 8 * GSTRIDE;      // input batch stride
constexpr int OD = 2 * D, OH = 2 * H, OW = 2 * W;
constexpr int WCHUNKS = W / 4;            // 12 float4-chunks per input row
constexpr int PER_BC  = D * H * WCHUNKS;  // 27648 threads per (b,c) slab
constexpr int BLOCK   = 256;
constexpr int GRIDX   = PER_BC / BLOCK;   // 108, exact

__global__ __launch_bounds__(BLOCK) void iwt3_kernel(const float* __restrict__ x,
                                                     float* __restrict__ out) {
    __shared__ f4 stage[8][BLOCK];        // 32 KB staging buffer

    const unsigned tid = threadIdx.x;
    const unsigned id  = blockIdx.x * BLOCK + tid;   // < 27648
    const unsigned wc  = id % WCHUNKS;
    const unsigned hh  = id / WCHUNKS;
    const unsigned h   = hh % H;
    const unsigned d   = hh / H;
    const unsigned c   = blockIdx.y;
    const unsigned b   = blockIdx.z;
    const unsigned w   = wc * 4;

    size_t in_base = (size_t)b * BSTRIDE + (size_t)c * CSTRIDE
                   + ((size_t)d * H + h) * W + w;

    // Stage the 8 wavelet-group tiles into LDS via the async engine.
    // Each instruction moves 32 lanes x 16 B. Low 32 bits of a flat __shared__
    // pointer are the LDS byte offset (hardware adds LDS_BASE).
#pragma unroll
    for (int g = 0; g < 8; ++g) {
        const float* gp = x + in_base + (size_t)g * GSTRIDE;
        unsigned lds_addr = (unsigned)(uintptr_t)(&stage[g][tid]);
        asm volatile("global_load_async_to_lds_b128 %0, %1, off"
                     :
                     : "v"(lds_addr), "v"(gp)
                     : "memory");
    }
    asm volatile("s_wait_asynccnt 0x0" ::: "memory");

    // Read back own 16B slots (no cross-lane sharing -> no barrier needed).
    f4 v[8];
#pragma unroll
    for (int g = 0; g < 8; ++g) v[g] = stage[g][tid];

    size_t ob = (((size_t)(b * C + c) * OD + 2 * d) * OH + 2 * h) * OW + 2 * w;

#pragma unroll
    for (int d0 = 0; d0 < 2; ++d0) {
        f4 x1 = v[4 * d0 + 0];
        f4 x2 = v[4 * d0 + 1];
        f4 x3 = v[4 * d0 + 2];
        f4 x4 = v[4 * d0 + 3];

        // Factored butterfly.
        f4 m = x1 - x3, n = x2 - x4;   // h0 = 0 pair
        f4 p = x1 + x3, q = x2 + x4;   // h0 = 1 pair
        f4 e00 = 0.5f * (m - n);       // (h0=0, w0=0)
        f4 e01 = 0.5f * (m + n);       // (h0=0, w0=1)
        f4 e10 = 0.5f * (p - q);       // (h0=1, w0=0)
        f4 e11 = 0.5f * (p + q);       // (h0=1, w0=1)

        // Interleave w-parities -> contiguous float4 stores.
        f4 s0 = __builtin_shufflevector(e00, e01, 0, 4, 1, 5);
        f4 s1 = __builtin_shufflevector(e00, e01, 2, 6, 3, 7);
        f4 s2 = __builtin_shufflevector(e10, e11, 0, 4, 1, 5);
        f4 s3 = __builtin_shufflevector(e10, e11, 2, 6, 3, 7);

        size_t r0 = ob + (size_t)d0 * ((size_t)OH * OW);  // row (2d+d0, 2h+0)
        size_t r1 = r0 + OW;                              // row (2d+d0, 2h+1)
        __builtin_nontemporal_store(s0, reinterpret_cast<f4*>(out + r0));
        __builtin_nontemporal_store(s1, reinterpret_cast<f4*>(out + r0 + 4));
        __builtin_nontemporal_store(s2, reinterpret_cast<f4*>(out + r1));
        __builtin_nontemporal_store(s3, reinterpret_cast<f4*>(out + r1 + 4));
    }
}

extern "C" void kernel_launch(void* const* d_in, const int* in_sizes, int n_in,
                              void* d_out, int out_size, void* d_ws, size_t ws_size,
                              hipStream_t stream) {
    (void)in_sizes; (void)n_in; (void)out_size; (void)d_ws; (void)ws_size;
    const float* x = reinterpret_cast<const float*>(d_in[0]);
    float* out     = reinterpret_cast<float*>(d_out);

    iwt3_kernel<<<dim3(GRIDX, C, B), dim3(BLOCK), 0, stream>>>(x, out);
}